// RIPUMining_45268955300479
// MI455X (gfx1250) — compile-verified
//
#include <hip/hip_runtime.h>

// ---------------------------------------------------------------------------
// RIPU mining for MI455X (gfx1250, wave32).
//
// Pipeline (all on `stream`):
//   K0: zero histogram + candidate counters (ws is not re-poisoned to zero).
//   K1: streaming fused entropy+argmax over the 160MB input (memory bound,
//       ~7us at 23.3 TB/s). Emits entropy f32 (8MB) + labels u8 (2MB) -> L2.
//   K2: 16x16 tiles (14x14 valid interior). Box-sum of entropy computed as
//       D = T * E * T with T = tridiagonal ones, via 8 chained
//       V_WMMA_F32_16X16X4_F32 per tile (LDS mediates layout conversion).
//       Impurity via window-multiplicity identity (81 compares + <=9 v_log).
//       Scores written + per-block LDS histogram flushed to global.
//   K3: per-image scan of 2048-bin histogram -> threshold bin T, need count.
//   K4: deterministic two-level sum of scores strictly above bin T +
//       candidate collection for bin == T.
//   K5: per-image exact bisection select inside bin T, final output [4] f32.
//
// Workspace layout (needs ~19.2 MiB; assumed available):
//   [0,      8MiB)  entropy f32
//   [8MiB,  10MiB)  labels u8
//   [10MiB, 18MiB)  scores f32
//   [18MiB, +32KiB) histogram 4*2048 int
//   [+32KiB,+48B )  thresh[4], need[4], candcnt[4]
//   [18MiB+64KiB, +32KiB) partials 8192 f32
//   [18MiB+128KiB, +1MiB) candidates 4*65536 f32
// ---------------------------------------------------------------------------

#define NUM_IMG 4
#define NUM_CLS 19
#define IMG_H 512
#define IMG_W 1024
#define HW_SZ (IMG_H * IMG_W) /* 1<<19 */
#define EPS_F 1e-6f
#define INV_LOG_C 0.3396232486f /* 1/ln(19) */
#define NBINS 2048
#define TOPK 256
#define CAND_CAP 65536
#define TILES_X 74 /* ceil(1024/14) */
#define TILES_Y 37 /* ceil(512/14)  */
#define TILES_PER_IMG (TILES_X * TILES_Y) /* 2738 */
#define TPI_PAD 2744                      /* padded to multiple of 8 waves */

typedef float v2f __attribute__((ext_vector_type(2)));
typedef float v8f __attribute__((ext_vector_type(8)));

__device__ __forceinline__ float band(int i, int j) {
  const int dd = i - j;
  return (dd <= 1 && dd >= -1) ? 1.0f : 0.0f;
}

// ---------------------------------------------------------------- K0: zero
__global__ __launch_bounds__(256) void k0_zero(int* __restrict__ hist,
                                               int* __restrict__ candcnt) {
  const int i = blockIdx.x * 256 + threadIdx.x;
  if (i < NUM_IMG * NBINS) hist[i] = 0;
  if (i < NUM_IMG) candcnt[i] = 0;
}

// ------------------------------------------------- K1: entropy + argmax
__global__ __launch_bounds__(256) void k1_entropy_argmax(
    const float* __restrict__ logit, float* __restrict__ ent,
    unsigned char* __restrict__ lab) {
  const int g = blockIdx.x * 256 + threadIdx.x; // one thread = 4 pixels
  const int b = g >> 17;                        // HW/4 = 131072 per image
  const int p = (g & 131071) << 2;
  const float* base = logit + (size_t)b * NUM_CLS * HW_SZ + p;

  float e0 = 0.f, e1 = 0.f, e2 = 0.f, e3 = 0.f;
  float m0 = -1.f, m1 = -1.f, m2 = -1.f, m3 = -1.f;
  int i0 = 0, i1 = 0, i2 = 0, i3 = 0;
#pragma unroll
  for (int c = 0; c < NUM_CLS; ++c) {
    if (c + 2 < NUM_CLS)
      __builtin_prefetch(base + (size_t)(c + 2) * HW_SZ, 0, 0);
    const float4 v = *(const float4*)(base + (size_t)c * HW_SZ);
    e0 -= v.x * __logf(v.x + EPS_F);
    e1 -= v.y * __logf(v.y + EPS_F);
    e2 -= v.z * __logf(v.z + EPS_F);
    e3 -= v.w * __logf(v.w + EPS_F);
    if (v.x > m0) { m0 = v.x; i0 = c; }
    if (v.y > m1) { m1 = v.y; i1 = c; }
    if (v.z > m2) { m2 = v.z; i2 = c; }
    if (v.w > m3) { m3 = v.w; i3 = c; }
  }
  float4 eo;
  eo.x = e0 * INV_LOG_C;
  eo.y = e1 * INV_LOG_C;
  eo.z = e2 * INV_LOG_C;
  eo.w = e3 * INV_LOG_C;
  *(float4*)(ent + ((size_t)b << 19) + p) = eo;
  uchar4 lo;
  lo.x = (unsigned char)i0;
  lo.y = (unsigned char)i1;
  lo.z = (unsigned char)i2;
  lo.w = (unsigned char)i3;
  *(uchar4*)(lab + ((size_t)b << 19) + p) = lo;
}

// ------------------------- K2: WMMA box-sum + impurity + score + histogram
__global__ __launch_bounds__(256) void k2_score(
    const float* __restrict__ ent, const unsigned char* __restrict__ lab,
    float* __restrict__ scores, int* __restrict__ hist) {
  __shared__ float lds_f[8][512]; // [0..255] = E tile, [256..511] = F tile
  __shared__ unsigned char lds_lab[8][256];
  __shared__ int lds_hist[NBINS];

  const int tid = threadIdx.x;
  const int lane = tid & 31;
  const int widx = tid >> 5;
  const int hh = lane >> 4; // lane half (0 or 1)
  const int m = lane & 15;

  for (int i = tid; i < NBINS; i += 256) lds_hist[i] = 0;
  __syncthreads();

  const int tile = blockIdx.x * 8 + widx;
  const int b = tile / TPI_PAD;
  const int lt = tile - b * TPI_PAD;
  const bool active = (b < NUM_IMG) && (lt < TILES_PER_IMG);
  const int ty = lt / TILES_X;
  const int tx = lt - ty * TILES_X;
  const int gr0 = ty * 14 - 1;
  const int gc0 = tx * 14 - 1;

  // stage 16x16 entropy (zero-padded) + labels (0xFF-padded) into LDS
#pragma unroll
  for (int t = 0; t < 8; ++t) {
    const int idx = lane + 32 * t;
    const int row = idx >> 4;
    const int col = idx & 15;
    const int gr = gr0 + row;
    const int gc = gc0 + col;
    const bool inb = active && ((unsigned)gr < (unsigned)IMG_H) &&
                     ((unsigned)gc < (unsigned)IMG_W);
    const size_t gix = ((size_t)b << 19) + ((size_t)gr << 10) + (size_t)gc;
    lds_f[widx][idx] = inb ? ent[gix] : 0.0f;
    lds_lab[widx][idx] = inb ? lab[gix] : (unsigned char)0xFF;
  }

  // ---- step 1: F = T * E  (4 chained f32 WMMAs, K tiled by 4) ----
  v8f f = {};
#pragma unroll
  for (int k = 0; k < 4; ++k) {
    const int k0 = 4 * k + hh * 2;
    v2f a; // A-layout slice of tridiagonal T: rows M=m, cols k0,k0+1
    a.x = band(m, k0);
    a.y = band(m, k0 + 1);
    v2f bb; // B-layout slice of E: rows K=k0,k0+1, col N=m (LDS gather)
    bb.x = lds_f[widx][k0 * 16 + m];
    bb.y = lds_f[widx][(k0 + 1) * 16 + m];
    f = __builtin_amdgcn_wmma_f32_16x16x4_f32(false, a, false, bb, (short)0, f,
                                              false, false);
  }
  // spill F (C/D layout) so step 2 can regather it in A layout
#pragma unroll
  for (int r = 0; r < 8; ++r)
    lds_f[widx][256 + (r + hh * 8) * 16 + m] = f[r];

  // ---- step 2: D = F * T ----
  v8f d = {};
#pragma unroll
  for (int k = 0; k < 4; ++k) {
    const int k0 = 4 * k + hh * 2;
    const v2f a = *(const v2f*)&lds_f[widx][256 + m * 16 + k0];
    v2f bb;
    bb.x = band(k0, m);
    bb.y = band(k0 + 1, m);
    d = __builtin_amdgcn_wmma_f32_16x16x4_f32(false, a, false, bb, (short)0, d,
                                              false, false);
  }

  // ---- impurity + score for the 14x14 valid interior ----
#pragma unroll
  for (int r = 0; r < 8; ++r) {
    const int li = r + hh * 8;
    const int lj = m;
    if (li < 1 || li > 14 || lj < 1 || lj > 14) continue;
    const int gr = gr0 + li;
    const int gc = gc0 + lj;
    if (!active || (unsigned)gr >= (unsigned)IMG_H ||
        (unsigned)gc >= (unsigned)IMG_W)
      continue;
    int labs[9];
#pragma unroll
    for (int di = 0; di < 3; ++di)
#pragma unroll
      for (int dj = 0; dj < 3; ++dj)
        labs[di * 3 + dj] =
            (int)lds_lab[widx][(li + di - 1) * 16 + (lj + dj - 1)];
    int count = 0;
#pragma unroll
    for (int p = 0; p < 9; ++p) count += (labs[p] != 255) ? 1 : 0;
    const float invCount = 1.0f / (float)count;
    float acc = 0.0f;
#pragma unroll
    for (int p = 0; p < 9; ++p) {
      if (labs[p] == 255) continue;
      int n = 0;
#pragma unroll
      for (int q = 0; q < 9; ++q) n += (labs[q] == labs[p]) ? 1 : 0;
      acc += __logf((float)n * invCount + EPS_F);
    }
    const float ri = -acc * invCount * INV_LOG_C;
    const float ru = d[r] * (1.0f / 9.0f);
    const float s = ru * ri;
    scores[((size_t)b << 19) + ((size_t)gr << 10) + (size_t)gc] = s;
    int bin = (int)(s * (float)NBINS);
    bin = bin < 0 ? 0 : (bin > NBINS - 1 ? NBINS - 1 : bin);
    atomicAdd(&lds_hist[bin], 1);
  }
  __syncthreads();
  const int img = (blockIdx.x * 8) / TPI_PAD; // TPI_PAD % 8 == 0
  for (int i = tid; i < NBINS; i += 256) {
    const int v = lds_hist[i];
    if (v) atomicAdd(&hist[img * NBINS + i], v);
  }
}

// ---------------------------------------------- K3: threshold bin per image
__global__ void k3_threshold(const int* __restrict__ hist,
                             int* __restrict__ thresh,
                             int* __restrict__ needv) {
  const int b = threadIdx.x;
  if (b >= NUM_IMG) return;
  int cum = 0, T = -1, need = 0;
  for (int t = NBINS - 1; t >= 0; --t) {
    const int h = hist[b * NBINS + t];
    if (cum + h >= TOPK) { T = t; need = TOPK - cum; break; }
    cum += h;
  }
  if (T < 0) need = TOPK - cum; // degenerate: fewer than TOPK scores
  thresh[b] = T;
  needv[b] = need;
}

// ------------- K4: deterministic partial sums (bins > T) + candidate gather
__global__ __launch_bounds__(256) void k4_gather(
    const float* __restrict__ scores, const int* __restrict__ thresh,
    float* __restrict__ partials, float* __restrict__ cand,
    int* __restrict__ candcnt) {
  __shared__ float red[256];
  const int tid = threadIdx.x;
  const int g = blockIdx.x * 256 + tid;
  const int b = g >> 19;
  const float s = scores[g];
  const int T = thresh[b];
  int bin = (int)(s * (float)NBINS);
  bin = bin < 0 ? 0 : (bin > NBINS - 1 ? NBINS - 1 : bin);
  float val = (bin > T) ? s : 0.0f;
  if (bin == T) {
    const int i = atomicAdd(&candcnt[b], 1);
    if (i < CAND_CAP) cand[b * CAND_CAP + i] = s;
  }
  red[tid] = val;
  __syncthreads();
  for (int o = 128; o > 0; o >>= 1) {
    if (tid < o) red[tid] += red[tid + o];
    __syncthreads();
  }
  if (tid == 0) partials[blockIdx.x] = red[0];
}

// --------------------- K5: per-image exact select inside bin T, final output
__global__ __launch_bounds__(256) void k5_finish(
    const float* __restrict__ partials, const float* __restrict__ cand,
    const int* __restrict__ candcnt, const int* __restrict__ thresh,
    const int* __restrict__ needv, float* __restrict__ out) {
  __shared__ float fr[256];
  __shared__ int ir[256];
  const int b = blockIdx.x;
  const int tid = threadIdx.x;
  const int blocks_per_img = (NUM_IMG * HW_SZ / 256) / NUM_IMG; // 2048

  float acc = 0.0f;
  for (int k = tid; k < blocks_per_img; k += 256)
    acc += partials[b * blocks_per_img + k];
  fr[tid] = acc;
  __syncthreads();
  for (int o = 128; o > 0; o >>= 1) {
    if (tid < o) fr[tid] += fr[tid + o];
    __syncthreads();
  }
  float result = fr[0];
  __syncthreads();

  const int T = thresh[b];
  const int need = needv[b];
  int Nc = candcnt[b];
  if (Nc > CAND_CAP) Nc = CAND_CAP;

  if (need > 0 && Nc > 0 && T >= 0) {
    const float* cv = cand + b * CAND_CAP;
    float lo = (float)(T - 1) * (1.0f / (float)NBINS); // all cands > lo
    float hi = (float)(T + 1) * (1.0f / (float)NBINS); // no cand  > hi
    for (int it = 0; it < 64; ++it) {
      const float mid = 0.5f * (lo + hi);
      int c = 0;
      for (int i = tid; i < Nc; i += 256) c += (cv[i] > mid) ? 1 : 0;
      ir[tid] = c;
      __syncthreads();
      for (int o = 128; o > 0; o >>= 1) {
        if (tid < o) ir[tid] += ir[tid + o];
        __syncthreads();
      }
      const int ctot = ir[0];
      __syncthreads();
      if (ctot >= need) lo = mid; else hi = mid;
    }
    int c = 0;
    float s = 0.0f;
    for (int i = tid; i < Nc; i += 256) {
      const float v = cv[i];
      if (v > hi) { c += 1; s += v; }
    }
    ir[tid] = c;
    fr[tid] = s;
    __syncthreads();
    for (int o = 128; o > 0; o >>= 1) {
      if (tid < o) { ir[tid] += ir[tid + o]; fr[tid] += fr[tid + o]; }
      __syncthreads();
    }
    int take = need - ir[0];
    if (take < 0) take = 0;
    result += fr[0] + (float)take * hi; // tied values at the cutoff == hi
  }
  if (tid == 0) out[b] = result;
}

// ---------------------------------------------------------------------------
extern "C" void kernel_launch(void* const* d_in, const int* in_sizes, int n_in,
                              void* d_out, int out_size, void* d_ws,
                              size_t ws_size, hipStream_t stream) {
  const float* logit = (const float*)d_in[0];
  float* out = (float*)d_out;
  char* ws = (char*)d_ws;

  const size_t MiB = 1024ull * 1024ull;
  float* ent = (float*)(ws + 0);
  unsigned char* lab = (unsigned char*)(ws + 8 * MiB);
  float* scores = (float*)(ws + 10 * MiB);
  int* hist = (int*)(ws + 18 * MiB);
  int* thresh = (int*)(ws + 18 * MiB + 32768);
  int* needv = thresh + NUM_IMG;
  int* candcnt = thresh + 2 * NUM_IMG;
  float* partials = (float*)(ws + 18 * MiB + 64 * 1024);
  float* cand = (float*)(ws + 18 * MiB + 128 * 1024);

  k0_zero<<<(NUM_IMG * NBINS + 255) / 256, 256, 0, stream>>>(hist, candcnt);
  k1_entropy_argmax<<<(NUM_IMG * HW_SZ / 4) / 256, 256, 0, stream>>>(logit, ent,
                                                                     lab);
  k2_score<<<(NUM_IMG * TPI_PAD) / 8, 256, 0, stream>>>(ent, lab, scores, hist);
  k3_threshold<<<1, 32, 0, stream>>>(hist, thresh, needv);
  k4_gather<<<(NUM_IMG * HW_SZ) / 256, 256, 0, stream>>>(scores, thresh,
                                                         partials, cand,
                                                         candcnt);
  k5_finish<<<NUM_IMG, 256, 0, stream>>>(partials, cand, candcnt, thresh, needv,
                                         out);
}